// PEERBlock_58437325029377
// MI455X (gfx1250) — compile-verified
//
#include <hip/hip_runtime.h>
#include <hip/hip_bf16.h>
#include <math.h>
#include <stdint.h>

// ---------------------------------------------------------------------------
// PEER block for MI455X (gfx1250, wave32).
//   B=2, T=512, D=1024  -> N = 1024 tokens
//   H=8, KDIM=128, HALF=64, NKEYS=256, K=8
// Dominant cost: 805 MB of random expert-row gathers (~35us @ 23.3 TB/s).
// GEMMs (<3 GFLOP) run on V_WMMA_F32_16X16X4_F32 (fp32 in / fp32 accum).
// CDNA5 paths used: v_wmma_f32_16x16x4_f32, global_load_async_to_lds_b128
// (+ s_wait_asynccnt), global_prefetch_b8, ds_add_f32 LDS reduction.
// ---------------------------------------------------------------------------

typedef float v2f __attribute__((ext_vector_type(2)));
typedef float v8f __attribute__((ext_vector_type(8)));

#define NTOK   1024      // B*T
#define DMODEL 1024
#define NHEAD  8
#define NKEYS  256
#define HALF   64
#define TOPK   8

// ---------------------------------------------------------------------------
// Kernel 1: q = x @ Wq^T + bq      [1024 x 1024] = [1024 x 1024][1024 x 1024]^T
// One wave computes a 16(M) x 64(N) strip: A fragment reused across 4 WMMAs.
// 1024 waves total. fp32 WMMA: A is 16x4, lanes 0-15 carry K={0,1},
// lanes 16-31 carry K={2,3}; B symmetric with N in lane[3:0].
// ---------------------------------------------------------------------------
__global__ void k_qproj(const float* __restrict__ x,
                        const float* __restrict__ Wq,
                        const float* __restrict__ bq,
                        float* __restrict__ q) {
  const int wid  = (blockIdx.x * blockDim.x + threadIdx.x) >> 5;
  const int lane = threadIdx.x & 31;
  const int mt = wid >> 4;                 // 64 row tiles (x rows)
  const int ns = wid & 15;                 // 16 col strips of 64 (Wq rows)
  const int lrow = lane & 15;
  const int kh = (lane >> 4) << 1;         // 0 (lanes 0-15) or 2 (lanes 16-31)

  const float* xrow = x + (size_t)(mt * 16 + lrow) * DMODEL;
  const float* wrow = Wq + (size_t)(ns * 64 + lrow) * DMODEL;

  v8f acc[4];
#pragma unroll
  for (int t = 0; t < 4; ++t) acc[t] = (v8f){0.f,0.f,0.f,0.f,0.f,0.f,0.f,0.f};

  for (int k = 0; k < DMODEL; k += 4) {
    v2f a = *(const v2f*)(xrow + k + kh);
#pragma unroll
    for (int t = 0; t < 4; ++t) {
      v2f b = *(const v2f*)(wrow + (size_t)t * 16 * DMODEL + k + kh);
      acc[t] = __builtin_amdgcn_wmma_f32_16x16x4_f32(
          false, a, false, b, (short)0, acc[t], false, false);
    }
  }

  // C/D layout: VGPR r -> M = r + 8*(lane>=16), N = lane & 15
  const int mrow = mt * 16 + ((lane >> 4) << 3);
  const int col0 = ns * 64 + (lane & 15);
#pragma unroll
  for (int t = 0; t < 4; ++t) {
    const int col = col0 + t * 16;
    const float bias = bq[col];
#pragma unroll
    for (int r = 0; r < 8; ++r) {
      q[(size_t)(mrow + r) * DMODEL + col] = acc[t][r] + bias;
    }
  }
}

// ---------------------------------------------------------------------------
// Kernel 2: sub-key scores.  For hp = h*2+p (16 combos):
//   s[n, hp, kk] = sum_c q[n, hp*64 + c] * keys[hp, kk, c],  c < 64
// One wave computes 16(n) x 64(kk); 4096 waves. K loop = 16 WMMA steps.
// ---------------------------------------------------------------------------
__global__ void k_score(const float* __restrict__ q,
                        const float* __restrict__ keys,
                        float* __restrict__ s) {
  const int wid  = (blockIdx.x * blockDim.x + threadIdx.x) >> 5;
  const int lane = threadIdx.x & 31;
  const int ns = wid & 3;                  // 4 strips of 64 keys
  const int mt = (wid >> 2) & 63;          // 64 token tiles
  const int hp = wid >> 8;                 // 0..15 = h*2 + p
  const int lrow = lane & 15;
  const int kh = (lane >> 4) << 1;

  // q[n, h*128 + p*64 + c] == q[n, hp*64 + c]
  const float* qrow = q + (size_t)(mt * 16 + lrow) * DMODEL + hp * HALF;
  const float* krow = keys + (size_t)(hp * NKEYS + ns * 64 + lrow) * HALF;

  v8f acc[4];
#pragma unroll
  for (int t = 0; t < 4; ++t) acc[t] = (v8f){0.f,0.f,0.f,0.f,0.f,0.f,0.f,0.f};

  for (int k = 0; k < HALF; k += 4) {
    v2f a = *(const v2f*)(qrow + k + kh);
#pragma unroll
    for (int t = 0; t < 4; ++t) {
      v2f b = *(const v2f*)(krow + (size_t)t * 16 * HALF + k + kh);
      acc[t] = __builtin_amdgcn_wmma_f32_16x16x4_f32(
          false, a, false, b, (short)0, acc[t], false, false);
    }
  }

  const int mrow = mt * 16 + ((lane >> 4) << 3);
  const int col0 = ns * 64 + (lane & 15);
#pragma unroll
  for (int t = 0; t < 4; ++t) {
    const int col = col0 + t * 16;
#pragma unroll
    for (int r = 0; r < 8; ++r) {
      s[((size_t)(mrow + r) * 16 + hp) * NKEYS + col] = acc[t][r];
    }
  }
}

// ---------------------------------------------------------------------------
// Kernel 3: per-(token, head) product-key top-k.
//   top-8 of s1[256], top-8 of s2[256]  (stable: ties keep lower index,
//   matching jax.lax.top_k), 8x8 cartesian sums -> second output
//   (all_scores, [8192 x 64]), top-8 of the 64, softmax -> expert weights.
// One thread per (n, h); 8192 threads.
// ---------------------------------------------------------------------------
__device__ __forceinline__ void top8_256(const float* __restrict__ sv,
                                         float* sc, int* id) {
#pragma unroll
  for (int i = 0; i < TOPK; ++i) { sc[i] = -3.4028235e38f; id[i] = 0; }
  for (int k = 0; k < NKEYS; ++k) {
    const float v = sv[k];
    if (v > sc[TOPK - 1]) {
      int pos = TOPK - 1;
      while (pos > 0 && sc[pos - 1] < v) {
        sc[pos] = sc[pos - 1]; id[pos] = id[pos - 1]; --pos;
      }
      sc[pos] = v; id[pos] = k;
    }
  }
}

__global__ void k_topk(const float* __restrict__ s,
                       float* __restrict__ all_scores,   // [8192 x 64]
                       int* __restrict__ eidx,           // [8192 x 8]
                       float* __restrict__ ewgt) {       // [8192 x 8]
  const int t = blockIdx.x * blockDim.x + threadIdx.x;   // n*8 + h
  if (t >= NTOK * NHEAD) return;

  const float* s1 = s + (size_t)t * 2 * NKEYS;
  const float* s2 = s1 + NKEYS;

  float sc1[TOPK], sc2[TOPK];
  int   i1[TOPK],  i2[TOPK];
  top8_256(s1, sc1, i1);
  top8_256(s2, sc2, i2);

  float csc[TOPK]; int cid[TOPK];
#pragma unroll
  for (int i = 0; i < TOPK; ++i) { csc[i] = -3.4028235e38f; cid[i] = 0; }

  float* o = all_scores + (size_t)t * (TOPK * TOPK);
  for (int i = 0; i < TOPK; ++i) {
    for (int j = 0; j < TOPK; ++j) {
      const float v = sc1[i] + sc2[j];
      o[i * TOPK + j] = v;
      if (v > csc[TOPK - 1]) {
        const int ei = i1[i] * NKEYS + i2[j];
        int pos = TOPK - 1;
        while (pos > 0 && csc[pos - 1] < v) {
          csc[pos] = csc[pos - 1]; cid[pos] = cid[pos - 1]; --pos;
        }
        csc[pos] = v; cid[pos] = ei;
      }
    }
  }

  // softmax over the selected 8 (csc sorted descending -> csc[0] is max)
  const float m = csc[0];
  float e[TOPK], sum = 0.f;
#pragma unroll
  for (int k = 0; k < TOPK; ++k) { e[k] = expf(csc[k] - m); sum += e[k]; }
  const float inv = 1.f / sum;
#pragma unroll
  for (int k = 0; k < TOPK; ++k) {
    eidx[t * TOPK + k] = cid[k];
    ewgt[t * TOPK + k] = e[k] * inv;
  }
}

// ---------------------------------------------------------------------------
// Kernel 4: expert gather + GLU + weighted combine. THE bandwidth stage.
// One block per token (256 threads = 8 waves). Wave w owns 8 of the 64
// (head, k) expert slots. x is staged into LDS with CDNA5 async b128 loads
// (no VGPR round-trip). While computing expert s, each wave prefetches the
// next expert's three 4KB rows (one global_prefetch_b8 per matrix: 32 lanes
// x 1 cacheline = whole row). Dots use float4 coalesced reads + wave
// shuffle-reduce; h = relu(x.wd)*(x.wg)*softmax_w; output partials live in
// registers and are combined across waves with ds_add_f32.
// ---------------------------------------------------------------------------
__global__ void k_expert(const float* __restrict__ x,
                         const float* __restrict__ w_up,
                         const float* __restrict__ w_down,
                         const float* __restrict__ w_gate,
                         const int* __restrict__ eidx,
                         const float* __restrict__ ewgt,
                         float* __restrict__ out) {
  __shared__ float smx[DMODEL];
  __shared__ float sacc[DMODEL];

  const int n   = blockIdx.x;
  const int tid = threadIdx.x;

  // --- async global->LDS staging of x[n] (256 lanes x 16B = 4KB) ---------
  {
    // Low 32 bits of the flat address of a __shared__ object are the LDS
    // byte offset (aperture bits live in [63:32]).
    const uint32_t lds_off = (uint32_t)(uintptr_t)(&smx[0]) + (uint32_t)tid * 16u;
    const uint64_t gaddr   = (uint64_t)(x + (size_t)n * DMODEL + tid * 4);
    asm volatile("global_load_async_to_lds_b128 %0, %1, off"
                 :: "v"(lds_off), "v"(gaddr) : "memory");
  }
  ((float4*)sacc)[tid] = make_float4(0.f, 0.f, 0.f, 0.f);
  asm volatile("s_wait_asynccnt 0x0" ::: "memory");
  __syncthreads();

  const int wave = tid >> 5;
  const int lane = tid & 31;

  float racc[32];
#pragma unroll
  for (int i = 0; i < 32; ++i) racc[i] = 0.f;

  for (int s8 = 0; s8 < 8; ++s8) {
    const int slot = wave * 8 + s8;               // (h*8 + k) flat
    const int   idx = eidx[n * 64 + slot];
    const float wgt = ewgt[n * 64 + slot];
    const float* wdp = w_down + (size_t)idx * DMODEL;
    const float* wgp = w_gate + (size_t)idx * DMODEL;

    // Prefetch next expert's rows: lane l pulls cacheline l (32 x 128B = 4KB).
    if (s8 < 7) {
      const int nidx = eidx[n * 64 + slot + 1];
      const size_t nb = (size_t)nidx * DMODEL + lane * 32;   // 32 floats = 128B
      __builtin_prefetch(w_down + nb, 0, 1);
      __builtin_prefetch(w_gate + nb, 0, 1);
      __builtin_prefetch(w_up   + nb, 0, 1);
    }

    float dd = 0.f, dg = 0.f;
#pragma unroll
    for (int j = 0; j < 8; ++j) {
      const int d = j * 128 + lane * 4;
      const float4 xd = *(const float4*)(smx + d);
      const float4 vd = *(const float4*)(wdp + d);
      const float4 vg = *(const float4*)(wgp + d);
      dd += xd.x * vd.x + xd.y * vd.y + xd.z * vd.z + xd.w * vd.w;
      dg += xd.x * vg.x + xd.y * vg.y + xd.z * vg.z + xd.w * vg.w;
    }
#pragma unroll
    for (int off = 16; off > 0; off >>= 1) {
      dd += __shfl_xor(dd, off, 32);
      dg += __shfl_xor(dg, off, 32);
    }
    const float h = fmaxf(dd, 0.f) * dg * wgt;

    const float* wup = w_up + (size_t)idx * DMODEL;
#pragma unroll
    for (int j = 0; j < 8; ++j) {
      const int d = j * 128 + lane * 4;
      const float4 vu = *(const float4*)(wup + d);
      racc[j * 4 + 0] += h * vu.x;
      racc[j * 4 + 1] += h * vu.y;
      racc[j * 4 + 2] += h * vu.z;
      racc[j * 4 + 3] += h * vu.w;
    }
  }

#pragma unroll
  for (int j = 0; j < 8; ++j) {
    const int d = j * 128 + lane * 4;
    atomicAdd(&sacc[d + 0], racc[j * 4 + 0]);
    atomicAdd(&sacc[d + 1], racc[j * 4 + 1]);
    atomicAdd(&sacc[d + 2], racc[j * 4 + 2]);
    atomicAdd(&sacc[d + 3], racc[j * 4 + 3]);
  }
  __syncthreads();

  ((float4*)(out + (size_t)n * DMODEL))[tid] = ((float4*)sacc)[tid];
}

// ---------------------------------------------------------------------------
// Launch. Inputs: hidden_states, Wq, bq, keys, w_up, w_down, w_gate (all f32).
// Output: out [2,512,1024] (1,048,576 f32) ++ all_scores [8192,64] (524,288).
// ---------------------------------------------------------------------------
extern "C" void kernel_launch(void* const* d_in, const int* in_sizes, int n_in,
                              void* d_out, int out_size, void* d_ws, size_t ws_size,
                              hipStream_t stream) {
  const float* x      = (const float*)d_in[0];
  const float* Wq     = (const float*)d_in[1];
  const float* bq     = (const float*)d_in[2];
  const float* keys   = (const float*)d_in[3];
  const float* w_up   = (const float*)d_in[4];
  const float* w_down = (const float*)d_in[5];
  const float* w_gate = (const float*)d_in[6];

  float* out        = (float*)d_out;
  float* all_scores = out + (size_t)NTOK * DMODEL;          // +1,048,576

  // workspace layout (floats): q | s | eidx | ewgt
  float* q    = (float*)d_ws;
  float* s    = q + (size_t)NTOK * DMODEL;                  // 1,048,576
  int*   eidx = (int*)(s + (size_t)NTOK * NHEAD * 2 * NKEYS); // +4,194,304
  float* ewgt = (float*)(eidx + NTOK * NHEAD * TOPK);       // +65,536

  k_qproj <<<256,  128, 0, stream>>>(x, Wq, bq, q);
  k_score <<<1024, 128, 0, stream>>>(q, keys, s);
  k_topk  <<<32,   256, 0, stream>>>(s, all_scores, eidx, ewgt);
  k_expert<<<NTOK, 256, 0, stream>>>(x, w_up, w_down, w_gate, eidx, ewgt, out);
}